// GraphAttention_16647293239609
// MI455X (gfx1250) — compile-verified
//
#include <hip/hip_runtime.h>

// GAT layer for MI455X (gfx1250, wave32).
// Stage 1: h = feat @ W^T + bw via V_WMMA_F32_16X16X4_F32 (full f32 — GEMM is
//          ~1.6 GFLOP vs an ~80 MB / 3.5 us HBM floor, so no reason to drop
//          precision). Fused epilogue also emits p[n]=h[n].a_self, q[n]=h[n].a_nbr.
// Stage 2: edge logits from p/q, 16-wide softmax via wave shuffles, weighted
//          gather-sum of h rows (gather set is 25.6 MB -> fully L2-resident).

#define N_IN   128
#define N_OUT  128
#define DEG    16
#define NEG_SLOPE 0.01f

typedef __attribute__((ext_vector_type(2))) float v2f;
typedef __attribute__((ext_vector_type(4))) float v4f;
typedef __attribute__((ext_vector_type(8))) float v8f;

__global__ __launch_bounds__(256) void gat_gemm_h(
    const float* __restrict__ feat, const float* __restrict__ W,
    const float* __restrict__ bw,   const float* __restrict__ aw,
    float* __restrict__ h, float* __restrict__ p, float* __restrict__ q,
    int n_nodes)
{
  const int lane = threadIdx.x & 31;
  const int wave = threadIdx.x >> 5;
  const int row0 = blockIdx.x * 128 + wave * 16;   // 16 rows per wave
  if (row0 >= n_nodes) return;                     // wave-uniform; n % 16 == 0

  const int m    = lane & 15;                      // A: M index / B,C,D: N index
  const int half = lane >> 4;

  // A (16x4 f32 per WMMA): lanes 0-15 M=lane, VGPR e -> K = e + 2*half.
  // Hoist the full 16x128 feature slab: 32 K-steps of v2f each.
  const float* arow = feat + (size_t)(row0 + m) * N_IN + 2 * half;
  v2f A[32];
#pragma unroll
  for (int j = 0; j < 32; ++j) A[j] = *(const v2f*)(arow + 4 * j);

  float pacc[8], qacc[8];
#pragma unroll
  for (int r = 0; r < 8; ++r) { pacc[r] = 0.f; qacc[r] = 0.f; }

#pragma unroll
  for (int nt = 0; nt < 8; ++nt) {
    const int col = nt * 16 + m;                   // C/D: N = lane & 15
    const float bwc = bw[col];
    v8f acc;
#pragma unroll
    for (int r = 0; r < 8; ++r) acc[r] = bwc;      // bias folded into C input

    // B (4x16 f32, K x N): lane's N = lane&15, element e -> K = e + 2*half.
    // W is (OUT, IN) row-major, so h[m,n] = sum_k feat[m,k] * W[n,k].
    const float* brow = W + (size_t)col * N_IN + 2 * half;
#pragma unroll
    for (int j = 0; j < 32; ++j) {
      v2f B = *(const v2f*)(brow + 4 * j);
      acc = __builtin_amdgcn_wmma_f32_16x16x4_f32(false, A[j], false, B,
                                                  (short)0, acc, false, false);
    }

    const float a_self = aw[col];
    const float a_nbr  = aw[N_OUT + col];
#pragma unroll
    for (int r = 0; r < 8; ++r) {
      const int row = row0 + r + 8 * half;         // C/D: M = r + 8*half
      h[(size_t)row * N_OUT + col] = acc[r];
      pacc[r] += acc[r] * a_self;                  // fused attention projections
      qacc[r] += acc[r] * a_nbr;
    }
  }

  // Reduce p/q partials across the 16 lanes that share a row set (xor masks
  // 1,2,4,8 stay within each half of the wave32).
#pragma unroll
  for (int r = 0; r < 8; ++r) {
    float pv = pacc[r], qv = qacc[r];
#pragma unroll
    for (int mask = 1; mask < 16; mask <<= 1) {
      pv += __shfl_xor(pv, mask, 32);
      qv += __shfl_xor(qv, mask, 32);
    }
    if (m == 0) {
      const int row = row0 + r + 8 * half;
      p[row] = pv;
      q[row] = qv;
    }
  }
}

__global__ __launch_bounds__(256) void gat_attn(
    const int* __restrict__ nodes, const int* __restrict__ nbrs,
    const float* __restrict__ h,   const float* __restrict__ p,
    const float* __restrict__ q,   const float* __restrict__ ab,
    float* __restrict__ out, int n_nodes)
{
  const int t = blockIdx.x * 8 + (threadIdx.x >> 5);   // one wave per node
  if (t >= n_nodes) return;
  const int lane = threadIdx.x & 31;

  // Lanes 0-15 own edge k = lane; lanes 16-31 mirror (keeps shuffles trivial).
  const int   k  = lane & 15;
  const float pt = p[nodes[t]];
  const int   nb = nbrs[t * DEG + k];
  float e = pt + q[nb] + ab[0];
  e = (e >= 0.f) ? e : NEG_SLOPE * e;                  // leaky ReLU

  float mx = e;
#pragma unroll
  for (int mask = 1; mask < 16; mask <<= 1)
    mx = fmaxf(mx, __shfl_xor(mx, mask, 32));
  const float ex = __expf(e - mx);
  float s = ex;
#pragma unroll
  for (int mask = 1; mask < 16; mask <<= 1)
    s += __shfl_xor(s, mask, 32);
  const float alpha = ex / s;

  // out[t, c:c+4] = sum_k alpha_k * h[nbr_k, c:c+4]  (L2-resident gather)
  v4f acc = {0.f, 0.f, 0.f, 0.f};
  const int c = lane * 4;
#pragma unroll
  for (int kk = 0; kk < DEG; ++kk) {
    const float a  = __shfl(alpha, kk, 32);
    const int   nn = __shfl(nb,    kk, 32);
    const v4f hv = *(const v4f*)(h + (size_t)nn * N_OUT + c);
#pragma unroll
    for (int i = 0; i < 4; ++i) acc[i] += a * hv[i];
  }
  *(v4f*)(out + (size_t)t * N_OUT + c) = acc;
}

extern "C" void kernel_launch(void* const* d_in, const int* in_sizes, int n_in,
                              void* d_out, int out_size, void* d_ws, size_t ws_size,
                              hipStream_t stream) {
  const float* feat  = (const float*)d_in[0];
  const int*   nodes = (const int*)  d_in[1];
  const int*   nbrs  = (const int*)  d_in[2];
  const float* W     = (const float*)d_in[3];
  const float* bw    = (const float*)d_in[4];
  const float* aw    = (const float*)d_in[5];
  const float* ab    = (const float*)d_in[6];
  float*       out   = (float*)d_out;

  const int n = in_sizes[1];                 // N_NODES (nodes array length)

  float* h = (float*)d_ws;                   // n * 128 floats
  float* p = h + (size_t)n * N_OUT;          // n floats
  float* q = p + n;                          // n floats

  gat_gemm_h<<<(n + 127) / 128, 256, 0, stream>>>(feat, W, bw, aw, h, p, q, n);
  gat_attn  <<<(n + 7) / 8,     256, 0, stream>>>(nodes, nbrs, h, p, q, ab, out, n);
}